// Regressor_33028298506662
// MI455X (gfx1250) — compile-verified
//
#include <hip/hip_runtime.h>

#define N_NODES  50000
#define N_EDGES  600000
#define D_IN     128
#define H        64
#define N_GRAPHS 500

typedef __attribute__((ext_vector_type(2))) float v2f;
typedef __attribute__((ext_vector_type(8))) float v8f;

// ---- workspace layout (float offsets) ----
#define AGG_OFF   0
#define AGG_SZ    (N_NODES * D_IN)            // 6,400,000
#define CNT_OFF   (AGG_OFF + AGG_SZ)          // 6,400,000
#define CNT_SZ    (N_NODES)                   // 50,000
#define GSUM_OFF  (CNT_OFF + CNT_SZ)          // 6,450,000
#define GSUM_SZ   (N_GRAPHS * H)              // 32,000
#define GCNT_OFF  (GSUM_OFF + GSUM_SZ)        // 6,482,000
#define GCNT_SZ   (N_GRAPHS)                  // 500
#define ZERO_SZ   (GCNT_OFF + GCNT_SZ)        // 6,482,500 (contiguous region to zero)
#define X1_OFF    (ZERO_SZ)                   // 6,482,500
#define X1_SZ     (N_NODES * H)               // 3,200,000

__device__ __forceinline__ void atomic_add_f32(float* p, float v) {
    __hip_atomic_fetch_add(p, v, __ATOMIC_RELAXED, __HIP_MEMORY_SCOPE_AGENT);
}

// ---------------- K0: zero the accumulation regions ----------------
__global__ void zero_ws_kernel(float4* ws4, int n4) {
    int i = blockIdx.x * blockDim.x + threadIdx.x;
    if (i < n4) ws4[i] = make_float4(0.f, 0.f, 0.f, 0.f);
}

// ---------------- K1: edge scatter-add (segment_sum over edges) ----------------
// grid = N_EDGES blocks, 128 threads (one per feature)
__global__ void edge_scatter_kernel(const float* __restrict__ x,
                                    const int* __restrict__ src,
                                    const int* __restrict__ dst,
                                    float* __restrict__ agg,
                                    float* __restrict__ cnt) {
    int e = blockIdx.x;
    int s = src[e];
    int d = dst[e];
    int f = threadIdx.x;
    atomic_add_f32(&agg[(size_t)d * D_IN + f], x[(size_t)s * D_IN + f]);
    if (f == 0) atomic_add_f32(&cnt[d], 1.0f);
}

// ---------------- K2: fused SAGE linear via WMMA f32 16x16x4 ----------------
// x1 = (agg/max(cnt,1)) @ Wl + bl + x @ Wr          [N_NODES, 64]
// 256 threads = 8 waves; each wave computes a 16x64 output tile.
__global__ void sage_gemm_kernel(const float* __restrict__ x,
                                 const float* __restrict__ agg,
                                 const float* __restrict__ cnt,
                                 const float* __restrict__ Wl,
                                 const float* __restrict__ bl,
                                 const float* __restrict__ Wr,
                                 float* __restrict__ x1) {
    __shared__ float sWl[D_IN * H];   // 32 KB
    __shared__ float sWr[D_IN * H];   // 32 KB
    for (int i = threadIdx.x; i < D_IN * H; i += blockDim.x) {
        sWl[i] = Wl[i];
        sWr[i] = Wr[i];
    }
    __syncthreads();

    const int wave  = threadIdx.x >> 5;    // wave32
    const int lane  = threadIdx.x & 31;
    const int mtile = blockIdx.x * 8 + wave;
    if (mtile * 16 >= N_NODES) return;     // wave-uniform: EXEC stays all-ones

    const int rlo  = lane & 15;            // row within tile / output column index
    const int koff = (lane >> 4) * 2;      // A/B fragment K offset: lanes 16-31 hold K+2,K+3
    const int row  = mtile * 16 + rlo;

    float rc = 1.0f / fmaxf(cnt[row], 1.0f);
    const float* __restrict__ xrow = x   + (size_t)row * D_IN;
    const float* __restrict__ arow = agg + (size_t)row * D_IN;

    v8f acc0 = {0.f,0.f,0.f,0.f,0.f,0.f,0.f,0.f};
    v8f acc1 = acc0, acc2 = acc0, acc3 = acc0;

    #pragma unroll 4
    for (int kb = 0; kb < D_IN; kb += 4) {
        const int k = kb + koff;           // even -> 8B-aligned float2 loads
        v2f am = *(const v2f*)(arow + k);
        am.x *= rc; am.y *= rc;
        v2f ax = *(const v2f*)(xrow + k);

        #pragma unroll
        for (int n = 0; n < 4; ++n) {
            const int col = n * 16 + rlo;
            v2f bm, bx;
            bm.x = sWl[ k      * H + col];
            bm.y = sWl[(k + 1) * H + col];
            bx.x = sWr[ k      * H + col];
            bx.y = sWr[(k + 1) * H + col];
            v8f* acc = (n == 0) ? &acc0 : (n == 1) ? &acc1 : (n == 2) ? &acc2 : &acc3;
            *acc = __builtin_amdgcn_wmma_f32_16x16x4_f32(
                       false, am, false, bm, (short)0, *acc, false, false);
            *acc = __builtin_amdgcn_wmma_f32_16x16x4_f32(
                       false, ax, false, bx, (short)0, *acc, false, false);
        }
    }

    // Epilogue: + bl, store. D layout: VGPR j -> M = j + 8*(lane>=16), N = lane&15.
    const int mrow = mtile * 16 + (lane >> 4) * 8;
    v8f accs[4] = {acc0, acc1, acc2, acc3};
    #pragma unroll
    for (int n = 0; n < 4; ++n) {
        const int col  = n * 16 + rlo;
        const float bb = bl[col];
        #pragma unroll
        for (int j = 0; j < 8; ++j) {
            x1[(size_t)(mrow + j) * H + col] = accs[n][j] + bb;
        }
    }
}

// ---------------- K3: global mean pool scatter ----------------
// grid = N_NODES blocks, 64 threads (one per hidden feature)
__global__ void pool_scatter_kernel(const float* __restrict__ x1,
                                    const int* __restrict__ batch,
                                    float* __restrict__ gsum,
                                    float* __restrict__ gcnt) {
    int node = blockIdx.x;
    int g = batch[node];
    atomic_add_f32(&gsum[(size_t)g * H + threadIdx.x], x1[(size_t)node * H + threadIdx.x]);
    if (threadIdx.x == 0) atomic_add_f32(&gcnt[g], 1.0f);
}

// ---------------- K4: per-graph MLP head (64->32->16->8->1) ----------------
__global__ void mlp_head_kernel(const float* __restrict__ gsum,
                                const float* __restrict__ gcnt,
                                const float* __restrict__ W0, const float* __restrict__ b0,
                                const float* __restrict__ W1, const float* __restrict__ b1,
                                const float* __restrict__ W2, const float* __restrict__ b2,
                                const float* __restrict__ W3, const float* __restrict__ b3,
                                float* __restrict__ out) {
    int g = blockIdx.x * blockDim.x + threadIdx.x;
    if (g >= N_GRAPHS) return;
    float inv = 1.0f / fmaxf(gcnt[g], 1.0f);

    float h0[64];
    #pragma unroll
    for (int k = 0; k < 64; ++k) h0[k] = gsum[(size_t)g * H + k] * inv;

    float h1[32];
    for (int j = 0; j < 32; ++j) {
        float s = b0[j];
        #pragma unroll 8
        for (int k = 0; k < 64; ++k) s += h0[k] * W0[k * 32 + j];
        h1[j] = fmaxf(s, 0.f);
    }
    float h2[16];
    for (int j = 0; j < 16; ++j) {
        float s = b1[j];
        #pragma unroll 8
        for (int k = 0; k < 32; ++k) s += h1[k] * W1[k * 16 + j];
        h2[j] = fmaxf(s, 0.f);
    }
    float h3[8];
    for (int j = 0; j < 8; ++j) {
        float s = b2[j];
        #pragma unroll
        for (int k = 0; k < 16; ++k) s += h2[k] * W2[k * 8 + j];
        h3[j] = fmaxf(s, 0.f);
    }
    float s = b3[0];
    #pragma unroll
    for (int k = 0; k < 8; ++k) s += h3[k] * W3[k];
    out[g] = s;
}

extern "C" void kernel_launch(void* const* d_in, const int* in_sizes, int n_in,
                              void* d_out, int out_size, void* d_ws, size_t ws_size,
                              hipStream_t stream) {
    const float* x          = (const float*)d_in[0];
    const int*   edge_index = (const int*)d_in[1];   // [2, N_EDGES]
    const int*   batch      = (const int*)d_in[2];
    const float* Wl         = (const float*)d_in[3];
    const float* bl         = (const float*)d_in[4];
    const float* Wr         = (const float*)d_in[5];
    const float* W0         = (const float*)d_in[6];
    const float* b0         = (const float*)d_in[7];
    const float* W1         = (const float*)d_in[8];
    const float* b1         = (const float*)d_in[9];
    const float* W2         = (const float*)d_in[10];
    const float* b2         = (const float*)d_in[11];
    const float* W3         = (const float*)d_in[12];
    const float* b3         = (const float*)d_in[13];
    float* out = (float*)d_out;

    float* ws   = (float*)d_ws;
    float* agg  = ws + AGG_OFF;
    float* cnt  = ws + CNT_OFF;
    float* gsum = ws + GSUM_OFF;
    float* gcnt = ws + GCNT_OFF;
    float* x1   = ws + X1_OFF;

    // K0: zero agg/cnt/gsum/gcnt (contiguous, ZERO_SZ divisible by 4)
    {
        int n4 = ZERO_SZ / 4;
        int blk = 256;
        zero_ws_kernel<<<(n4 + blk - 1) / blk, blk, 0, stream>>>((float4*)ws, n4);
    }
    // K1: edge scatter
    edge_scatter_kernel<<<N_EDGES, D_IN, 0, stream>>>(
        x, edge_index, edge_index + N_EDGES, agg, cnt);
    // K2: fused SAGE linear (WMMA). 3125 M-tiles, 8 per block.
    {
        int mtiles = N_NODES / 16;                 // 3125
        int blocks = (mtiles + 7) / 8;             // 391
        sage_gemm_kernel<<<blocks, 256, 0, stream>>>(x, agg, cnt, Wl, bl, Wr, x1);
    }
    // K3: pool
    pool_scatter_kernel<<<N_NODES, H, 0, stream>>>(x1, batch, gsum, gcnt);
    // K4: MLP head
    mlp_head_kernel<<<(N_GRAPHS + 255) / 256, 256, 0, stream>>>(
        gsum, gcnt, W0, b0, W1, b1, W2, b2, W3, b3, out);
}